// MeanField_39539468927325
// MI455X (gfx1250) — compile-verified
//
#include <hip/hip_runtime.h>
#include <hip/hip_bf16.h>

// ---------------------------------------------------------------------------
// MeanField CRF loop for MI455X (gfx1250, wave32).
//   iter_kernel (x20)  : 3x3 per-pixel-weight stencil; tile staged into LDS via
//                        CDNA5 async global->LDS (ASYNCcnt) from a zero-bordered
//                        padded field (no bounds checks; borders = zero padding)
//   zero/init/precompute/finalize/valid: setup + epilogue
// Workspace ~28.4 MB -> fully L2-resident on MI455X (192 MB L2).
// No matmul structure (per-pixel weights) => WMMA not applicable; VALU FMAs.
// ---------------------------------------------------------------------------

namespace {
constexpr int   Bn   = 4;
constexpr int   H_   = 384;
constexpr int   W_   = 384;
constexpr int   HW_  = H_ * W_;          // 147456
constexpr int   WP_  = W_ + 2;           // 386
constexpr int   WPHP_ = WP_ * (H_ + 2);  // 148996
constexpr float INV_G = -4.9832890f;     // 1 / ln(9/11)

constexpr size_t KBUF_BYTES = (size_t)Bn * 9 * HW_ * sizeof(float);  // 21,233,664
constexpr size_t T_OFF      = KBUF_BYTES;
constexpr size_t T_BYTES    = (size_t)Bn * HW_ * sizeof(float);
constexpr size_t B0_OFF     = T_OFF + T_BYTES;
constexpr size_t BP_BYTES   = (size_t)Bn * WPHP_ * sizeof(float);
constexpr size_t B1_OFF     = B0_OFF + BP_BYTES;
constexpr size_t CNT_OFF    = B1_OFF + BP_BYTES;                     // + 16 B
} // namespace

// ---- CDNA5 async global->LDS staging ---------------------------------------
#if defined(__has_builtin)
#if __has_builtin(__builtin_amdgcn_global_load_async_to_lds_b32)
#define MF_ASYNC_BUILTIN 1
#endif
#if __has_builtin(__builtin_amdgcn_s_wait_asynccnt)
#define MF_WAIT_BUILTIN 1
#endif
#endif

__device__ __forceinline__ void mf_async_copy_f32(const float* gsrc, float* lds_dst) {
#if defined(MF_ASYNC_BUILTIN)
  // Builtin signature (per clang diagnostic): (__device__ int*, __shared__ int*, Imm, Imm)
  __builtin_amdgcn_global_load_async_to_lds_b32(
      (__attribute__((address_space(1))) int*)gsrc,
      (__attribute__((address_space(3))) int*)lds_dst,
      /*offset=*/0, /*cpol=*/0);
#else
  unsigned laddr = (unsigned)(unsigned long long)(void*)lds_dst;
  asm volatile("global_load_async_to_lds_b32 %0, %1, off"
               :: "v"(laddr), "v"(gsrc) : "memory");
#endif
}

__device__ __forceinline__ void mf_wait_async0() {
#if defined(MF_WAIT_BUILTIN)
  __builtin_amdgcn_s_wait_asynccnt(0);
#else
  asm volatile("s_wait_asynccnt 0" ::: "memory");
#endif
}

// ---- hot kernel first (so the disasm snippet shows it) ----------------------

#define TX 128
#define TY 8

__global__ __launch_bounds__(256) void iter_kernel(const float* __restrict__ bIn,
                                                   float* __restrict__ bOut,
                                                   const float* __restrict__ kbuf,
                                                   const float* __restrict__ T) {
  __shared__ float tile[(TY + 2) * (TX + 2)];
  const int img = blockIdx.z;
  const int x0  = blockIdx.x * TX;   // image-space tile origin
  const int y0  = blockIdx.y * TY;
  const int tid = threadIdx.x;

  // Stage (TY+2)x(TX+2) halo tile of the zero-bordered padded field into LDS
  // via CDNA5 async global->LDS copies (ASYNCcnt). Padded layout => every
  // address is valid, no bounds checks, borders are the zero padding.
  const float* bsrc = bIn + (size_t)img * WPHP_;
  for (int i = tid; i < (TY + 2) * (TX + 2); i += 256) {
    int ly = i / (TX + 2);
    int lx = i - ly * (TX + 2);
    mf_async_copy_f32(bsrc + (size_t)(y0 + ly) * WP_ + (x0 + lx), &tile[i]);
  }
  mf_wait_async0();
  __syncthreads();

  const float* kb = kbuf + (size_t)img * 9 * HW_;
  const float* Tp = T + (size_t)img * HW_;
  float* bo = bOut + (size_t)img * WPHP_;

  const int lx     = tid & (TX - 1);
  const int lybase = (tid >> 7) * 4;   // 2 thread-rows x 4 pixels each
#pragma unroll
  for (int r = 0; r < 4; ++r) {
    int ly = lybase + r;
    int Y = y0 + ly, X = x0 + lx;
    int p = Y * W_ + X;
    float c = 0.f;
#pragma unroll
    for (int j = 0; j < 9; ++j) {
      int dy = j / 3 - 1, dx = j % 3 - 1;
      c += kb[(size_t)j * HW_ + p] * tile[(ly + 1 + dy) * (TX + 2) + (lx + 1 + dx)];
    }
    bo[(size_t)(Y + 1) * WP_ + (X + 1)] = (c > Tp[p]) ? 1.f : 0.f;
  }
}

// ---- setup / epilogue kernels ----------------------------------------------

__global__ void zero_kernel(float* b0, float* b1, int* counts) {
  int idx = blockIdx.x * blockDim.x + threadIdx.x;
  if (idx < Bn * WPHP_) { b0[idx] = 0.f; b1[idx] = 0.f; }
  if (idx < Bn) counts[idx] = 0;
}

__global__ void init_kernel(const float* __restrict__ x,
                            const float* __restrict__ tgt,
                            float* __restrict__ b0) {
  int idx = blockIdx.x * blockDim.x + threadIdx.x;
  if (idx >= Bn * HW_) return;
  int img = idx / HW_;
  int p   = idx - img * HW_;
  int y   = p / W_, xx = p - y * W_;
  float xm = x[idx] * tgt[idx];
  b0[(size_t)img * WPHP_ + (size_t)(y + 1) * WP_ + (xx + 1)] = (xm > 0.5f) ? 1.f : 0.f;
}

__global__ void precompute_kernel(const float* __restrict__ fm,
                                  const float* __restrict__ tgt,
                                  float* __restrict__ kbuf,
                                  float* __restrict__ T) {
  int idx = blockIdx.x * blockDim.x + threadIdx.x;
  if (idx >= Bn * HW_) return;
  int img = idx / HW_;
  int p   = idx - img * HW_;
  int y   = p / W_, x = p - y * W_;
  const float* f = fm + (size_t)img * 3 * HW_;
  float c0 = f[p] + 10.f;
  float c1 = f[HW_ + p] + 10.f;
  float c2 = f[2 * HW_ + p] + 10.f;
  // spatial[j] / (2*THETA1^2), j row-major over (dy,dx) in {-1,0,1}^2
  const float spd[9] = {2.f/1800.f, 1.f/1800.f, 2.f/1800.f,
                        1.f/1800.f, 0.f,        1.f/1800.f,
                        2.f/1800.f, 1.f/1800.f, 2.f/1800.f};
  float ksum = 0.f;
#pragma unroll
  for (int j = 0; j < 9; ++j) {
    int dy = j / 3 - 1, dx = j % 3 - 1;
    int yy = y + dy, xx = x + dx;
    float n0 = 0.f, n1 = 0.f, n2 = 0.f;       // zero padding of (fm+10), as in unfold
    if (yy >= 0 && yy < H_ && xx >= 0 && xx < W_) {
      int q = yy * W_ + xx;
      n0 = f[q] + 10.f; n1 = f[HW_ + q] + 10.f; n2 = f[2 * HW_ + q] + 10.f;
    }
    float d0 = n0 - c0, d1 = n1 - c1, d2 = n2 - c2;
    float color = -2.f * (d0 * d0 + d1 * d1 + d2 * d2);  // / (2*0.5^2)
    float w = 3.0f * expf(color - spd[j]);               // ALPHA0 * exp(...)
    kbuf[((size_t)img * 9 + j) * HW_ + p] = w;
    ksum += w;
  }
  // decision threshold: new_b = [ sum_j k_j*b_j > T ]  (exact reformulation)
  T[idx] = 0.5f * (ksum + logf(tgt[idx]) * INV_G);
}

__global__ void finalize_kernel(const float* __restrict__ bFin,
                                float* __restrict__ out,
                                int* __restrict__ counts) {
  __shared__ int sred[256];
  int idx = blockIdx.x * 256 + threadIdx.x;   // HW_ % 256 == 0 -> block in one image
  int v = 0;
  if (idx < Bn * HW_) {
    int img = idx / HW_;
    int p   = idx - img * HW_;
    int y   = p / W_, x = p - y * W_;
    float b = bFin[(size_t)img * WPHP_ + (size_t)(y + 1) * WP_ + (x + 1)];
    out[idx] = b;
    v = (b > 0.5f) ? 1 : 0;
  }
  sred[threadIdx.x] = v;
  __syncthreads();
  for (int s = 128; s > 0; s >>= 1) {
    if (threadIdx.x < s) sred[threadIdx.x] += sred[threadIdx.x + s];
    __syncthreads();
  }
  if (threadIdx.x == 0) atomicAdd(&counts[blockIdx.x / (HW_ / 256)], sred[0]);
}

__global__ void valid_kernel(const int* __restrict__ counts, float* __restrict__ validOut) {
  int b = threadIdx.x;
  if (b < Bn) {
    float cnt = (float)counts[b];
    validOut[b] = (cnt >= 0.05f * (float)HW_ && cnt <= 0.95f * (float)HW_) ? 1.f : 0.f;
  }
}

// ---------------------------------------------------------------------------

extern "C" void kernel_launch(void* const* d_in, const int* in_sizes, int n_in,
                              void* d_out, int out_size, void* d_ws, size_t ws_size,
                              hipStream_t stream) {
  const float* x   = (const float*)d_in[0];   // [4,1,384,384]
  const float* tgt = (const float*)d_in[1];   // [4,1,384,384]
  const float* fm  = (const float*)d_in[2];   // [4,3,384,384]
  float* out = (float*)d_out;                 // [4*384*384] out + [4] valid

  char*  ws     = (char*)d_ws;                // needs ~28.4 MB
  float* kbuf   = (float*)(ws);
  float* T      = (float*)(ws + T_OFF);
  float* b0     = (float*)(ws + B0_OFF);
  float* b1     = (float*)(ws + B1_OFF);
  int*   counts = (int*)(ws + CNT_OFF);

  const int npix   = Bn * HW_;
  const int npad   = Bn * WPHP_;

  zero_kernel<<<(npad + 255) / 256, 256, 0, stream>>>(b0, b1, counts);
  init_kernel<<<(npix + 255) / 256, 256, 0, stream>>>(x, tgt, b0);
  precompute_kernel<<<(npix + 255) / 256, 256, 0, stream>>>(fm, tgt, kbuf, T);

  float* bi = b0;
  float* bo = b1;
  for (int it = 0; it < 20; ++it) {
    iter_kernel<<<dim3(W_ / TX, H_ / TY, Bn), 256, 0, stream>>>(bi, bo, kbuf, T);
    float* tmp = bi; bi = bo; bo = tmp;
  }

  finalize_kernel<<<(npix + 255) / 256, 256, 0, stream>>>(bi, out, counts);
  valid_kernel<<<1, 32, 0, stream>>>(counts, out + npix);
}